// GMM_model_54597624267469
// MI455X (gfx1250) — compile-verified
//
#include <hip/hip_runtime.h>
#include <math.h>

typedef __attribute__((ext_vector_type(2))) float v2f;
typedef __attribute__((ext_vector_type(8))) float v8f;

#define IMG 512
#define NPIX (IMG * IMG)      // 262144
#define NB 8
#define NK 8
#define NQ 7
#define BLOCKS_PER_IMG 128
#define ITER1 8               // 128 blocks * 256 thr * 8 pix = 262144 pix/image
#define LOG_SQRT_2PI 0.91893853320467274178f

// d_ws float layout
#define WS_PART1 0                          // 1024 blocks * 56 floats
#define WS_MU    (1024 * 56)                // 192
#define WS_STD   (1024 * 56 + 192)          // 192
#define WS_LP    (1024 * 56 + 384)          // 1024
// d_out float layout: [0]=log_prob, [1..192]=mu, [193..384]=std, [385..]=gamma(B,H,W,K)
#define OUT_G 385

// ---------------------------------------------------------------------------
// Pass 1: softmax(gamma) + transposed gamma output + per-image moment GEMM
//   moments(8x7) = Gamma^T (8 x P) * W (P x 7), W = [1, D, h, s, D^2, h^2, s^2]
//   computed with V_WMMA_F32_16X16X4_F32, Kdim = 4 pixels per WMMA.
// ---------------------------------------------------------------------------
__global__ __launch_bounds__(256) void pass1_kernel(const float* __restrict__ z,
                                                    const float* __restrict__ x,
                                                    float* __restrict__ out,
                                                    float* __restrict__ ws) {
    __shared__ float sg[8][32][16];   // per-wave: gamma, rows=pixel, cols=k (padded to 16)
    __shared__ float swm[8][32][16];  // per-wave: W,     rows=pixel, cols=q (padded to 16)
    __shared__ float part[8][56];     // per-wave moment partials

    const int tid  = threadIdx.x;
    const int w    = tid >> 5;
    const int lane = tid & 31;
    const int half = lane >> 4;
    const int l16  = lane & 15;
    const int b    = blockIdx.x >> 7;        // image
    const int blk  = blockIdx.x & 127;
    const int pix0 = blk * (256 * ITER1);

    // zero the pad columns once (they are never rewritten)
#pragma unroll
    for (int j = 8; j < 16; ++j) { sg[w][lane][j] = 0.0f; swm[w][lane][j] = 0.0f; }

    const float* zb = z + (size_t)b * NK * NPIX;
    const float* xb = x + (size_t)b * 3 * NPIX;
    float* og = out + OUT_G + (size_t)b * NPIX * 8;

    v8f acc = {0.f, 0.f, 0.f, 0.f, 0.f, 0.f, 0.f, 0.f};

    for (int it = 0; it < ITER1; ++it) {
        const int pbase = pix0 + it * 256 + w * 32;   // wave's 32 contiguous pixels
        const int p = pbase + lane;

        // softmax over K (max-subtracted, like jax.nn.softmax)
        float zv[8];
#pragma unroll
        for (int k = 0; k < 8; ++k) zv[k] = zb[(size_t)k * NPIX + p];
        float mx = zv[0];
#pragma unroll
        for (int k = 1; k < 8; ++k) mx = fmaxf(mx, zv[k]);
        float se = 0.0f;
#pragma unroll
        for (int k = 0; k < 8; ++k) { zv[k] = __expf(zv[k] - mx); se += zv[k]; }
        const float inv = 1.0f / se;
#pragma unroll
        for (int k = 0; k < 8; ++k) { zv[k] *= inv; sg[w][lane][k] = zv[k]; }

        const float D = xb[p], h = xb[NPIX + p], s = xb[2 * NPIX + p];
        swm[w][lane][0] = 1.0f;
        swm[w][lane][1] = D;      swm[w][lane][2] = h;      swm[w][lane][3] = s;
        swm[w][lane][4] = D * D;  swm[w][lane][5] = h * h;  swm[w][lane][6] = s * s;
        __syncthreads();

        // coalesced transposed gamma store: wave writes 256 contiguous floats
        {
            float* dst = og + (size_t)pbase * 8;
#pragma unroll
            for (int j = 0; j < 8; ++j) {
                const int o = lane + j * 32;            // o = pix_in_wave*8 + k
                dst[o] = sg[w][o >> 3][o & 7];
            }
        }

        // 8 WMMAs over this wave's 32 pixels (Kdim = 4 pixels each).
        // A (16x4 f32): lanes 0-15 hold Kdim {0,1}, lanes 16-31 hold {2,3}; row m = k.
        // B (4x16 f32): mirrored Kdim split; col n = q.
#pragma unroll
        for (int c = 0; c < 8; ++c) {
            const int prow = c * 4 + half * 2;
            v2f a, bm;
            a.x  = sg[w][prow][l16];
            a.y  = sg[w][prow + 1][l16];
            bm.x = swm[w][prow][l16];
            bm.y = swm[w][prow + 1][l16];
            acc = __builtin_amdgcn_wmma_f32_16x16x4_f32(false, a, false, bm,
                                                        (short)0, acc, false, false);
        }
        __syncthreads();
    }

    // D layout: lane n (0..15), vgpr r -> D[row=r, col=n]. Valid: rows k=0..7, cols q=0..6.
    if (half == 0 && l16 < NQ) {
#pragma unroll
        for (int r = 0; r < 8; ++r) part[w][r * NQ + l16] = acc[r];
    }
    __syncthreads();
    if (tid < 56) {   // deterministic in-order combine of the 8 waves
        float sum = 0.0f;
#pragma unroll
        for (int ww = 0; ww < 8; ++ww) sum += part[ww][tid];
        ws[WS_PART1 + (size_t)blockIdx.x * 56 + tid] = sum;
    }
}

// ---------------------------------------------------------------------------
// Pass 2: reduce block partials (fixed order), derive mu/std (concat layout)
// ---------------------------------------------------------------------------
__global__ void pass2_kernel(float* __restrict__ out, float* __restrict__ ws) {
    const int t = threadIdx.x;     // 64 threads: one (b,k) each
    if (t >= 64) return;
    const int b = t >> 3, k = t & 7;
    float q[7];
#pragma unroll
    for (int j = 0; j < 7; ++j) {
        float s = 0.0f;
        for (int blk = 0; blk < BLOCKS_PER_IMG; ++blk)
            s += ws[WS_PART1 + (size_t)(b * BLOCKS_PER_IMG + blk) * 56 + k * 7 + j];
        q[j] = s;
    }
    const float S0 = q[0];
    const float S  = S0 + 1e-7f;
#pragma unroll
    for (int c = 0; c < 3; ++c) {
        const float G1 = q[1 + c], G2 = q[4 + c];
        const float M  = G1 / S;
        // sum_p gamma*(x-M)^2 == G2 - 2*M*G1 + M^2*S0
        const float wsum = G2 - 2.0f * M * G1 + M * M * S0;
        const float sd = sqrtf(fmaxf(wsum / S, 0.0f));
        const int idx = (c * 8 + b) * 8 + k;    // concat([M_d; M_a; M_b]) layout
        out[1 + idx]   = M;   ws[WS_MU + idx]  = M;
        out[193 + idx] = sd;  ws[WS_STD + idx] = sd;
    }
}

// ---------------------------------------------------------------------------
// Pass 3: per-pixel log-likelihood. NOTE: reference reshapes the channel-major
// concat as (B,3,K), so image b / channel c uses concatenated row (3b+c).
// ---------------------------------------------------------------------------
__global__ __launch_bounds__(256) void pass3_kernel(const float* __restrict__ x,
                                                    const float* __restrict__ out,
                                                    float* __restrict__ ws) {
    __shared__ float muL[3][8], aL[3][8], lL[3][8];
    __shared__ float red[256];
    const int tid = threadIdx.x;
    const int b   = blockIdx.x >> 7;
    const int blk = blockIdx.x & 127;
    const int pix0 = blk * 2048;

    if (tid < 24) {
        const int c = tid >> 3, k = tid & 7;
        const int row = 3 * b + c;               // reference's reshape quirk
        const float m = ws[WS_MU + row * 8 + k];
        const float s = ws[WS_STD + row * 8 + k];
        muL[c][k] = m;
        aL[c][k]  = -0.5f / (s * s);
        lL[c][k]  = -__logf(s);
    }
    __syncthreads();

    const float* xb = x + (size_t)b * 3 * NPIX;
    const float* og = out + OUT_G + (size_t)b * NPIX * 8;

    float accl = 0.0f;
    for (int j = 0; j < 8; ++j) {
        const int p = pix0 + j * 256 + tid;
        const float x0 = xb[p], x1 = xb[NPIX + p], x2 = xb[2 * NPIX + p];
        const float* g = og + (size_t)p * 8;
        float psum = 0.0f;
#pragma unroll
        for (int k = 0; k < 8; ++k) {
            const float d0 = x0 - muL[0][k];
            const float d1 = x1 - muL[1][k];
            const float d2 = x2 - muL[2][k];
            float t = aL[0][k] * d0 * d0 + lL[0][k]
                    + aL[1][k] * d1 * d1 + lL[1][k]
                    + aL[2][k] * d2 * d2 + lL[2][k];
            t = t * (1.0f / 3.0f) - LOG_SQRT_2PI;
            psum += __expf(t) * g[k];
        }
        const float prob = fminf(psum + 1e-7f, 1.0f);
        accl += __logf(prob);
    }
    red[tid] = accl;
    __syncthreads();
    for (int s = 128; s > 0; s >>= 1) {
        if (tid < s) red[tid] += red[tid + s];
        __syncthreads();
    }
    if (tid == 0) ws[WS_LP + blockIdx.x] = red[0];
}

// ---------------------------------------------------------------------------
// Pass 4: deterministic final reduction of 1024 block sums -> d_out[0]
// ---------------------------------------------------------------------------
__global__ void pass4_kernel(float* __restrict__ out, const float* __restrict__ ws) {
    __shared__ float red[256];
    const int t = threadIdx.x;
    red[t] = ws[WS_LP + t] + ws[WS_LP + t + 256] + ws[WS_LP + t + 512] + ws[WS_LP + t + 768];
    __syncthreads();
    for (int s = 128; s > 0; s >>= 1) {
        if (t < s) red[t] += red[t + s];
        __syncthreads();
    }
    if (t == 0) out[0] = red[0];
}

extern "C" void kernel_launch(void* const* d_in, const int* in_sizes, int n_in,
                              void* d_out, int out_size, void* d_ws, size_t ws_size,
                              hipStream_t stream) {
    const float* z = (const float*)d_in[0];   // (8, 8, 512, 512)
    const float* x = (const float*)d_in[1];   // (8, 3, 512, 512)
    float* out = (float*)d_out;
    float* ws  = (float*)d_ws;
    (void)in_sizes; (void)n_in; (void)out_size; (void)ws_size;

    pass1_kernel<<<NB * BLOCKS_PER_IMG, 256, 0, stream>>>(z, x, out, ws);
    pass2_kernel<<<1, 64, 0, stream>>>(out, ws);
    pass3_kernel<<<NB * BLOCKS_PER_IMG, 256, 0, stream>>>(x, out, ws);
    pass4_kernel<<<1, 256, 0, stream>>>(out, ws);
}